// TimeWarpingLayer_26809185862121
// MI455X (gfx1250) — compile-verified
//
#include <hip/hip_runtime.h>
#include <hip/hip_bf16.h>

// ---------------------------------------------------------------------------
// TimeWarpingLayer on MI455X (gfx1250, wave32).
//
// Shapes: T=1024, N=32, K = BANDS*C*FREQ = 1056, NK = 33792.
// Phase A: not-a-knot cubic spline forward Thomas scan along T. The
//   tridiagonal coefficients are data-independent, so the scan is blocked
//   into 16-step chunks: dp_chunk(16x16) = L_c(16x16) @ b_chunk + g_c*dp_in
//   -> four V_WMMA_F32_16X16X4_F32 per chunk (fp32), carry via ds_bpermute.
//   Chunks 0 and 63 are peeled so the 62 steady-state iterations have no
//   boundary branches (EXEC stays all-1s, as WMMA requires).
// Phase B: backward substitution, same chunked-WMMA structure in reverse,
//   overwriting dp with s in LDS.
// Phase C: Hermite warp per batch, gathering y from global (whole 138MB
//   tensor is L2-resident in the 192MB L2) and s from the wave's LDS strip.
//
// One wave (32 threads) owns a 16-column strip: s-strip = 1024*16*4B = 64KB
// LDS (320KB/WGP -> ~5 waves resident). Grid = 32 batches * 66 tiles = 2112.
// ---------------------------------------------------------------------------

typedef float v2f __attribute__((ext_vector_type(2)));
typedef float v8f __attribute__((ext_vector_type(8)));

#define TT 1024
#define NN 32
#define KK 1056
#define NKL 33792ll
#define NCHUNK 64

// workspace layout (floats)
#define WS_ALPHA 0
#define WS_BETA  1024
#define WS_GAMMA 2048
#define WS_GF    3072
#define WS_GB    4096
#define WS_AF    5120
#define WS_AB    21504   /* 5120 + 64*32*8 */

// ---------------------------------------------------------------------------
// K1: sequential tridiagonal coefficient recurrence (single thread).
// lower=[0,1..1,2], diag=[1,4..4,1], upper=[2,1..1,0]
// m_t = diag - lower*cp_prev ; cp = upper/m ; alpha=1/m ; beta=-lower/m ;
// gamma = -cp (backward substitution coefficient)
// ---------------------------------------------------------------------------
__global__ void tw_coef_kernel(float* ws) {
    float* alpha = ws + WS_ALPHA;
    float* beta  = ws + WS_BETA;
    float* gamma = ws + WS_GAMMA;
    float cp_prev = 0.0f;
    for (int t = 0; t < TT; ++t) {
        float lo = (t == 0) ? 0.0f : ((t == TT - 1) ? 2.0f : 1.0f);
        float di = (t == 0 || t == TT - 1) ? 1.0f : 4.0f;
        float up = (t == 0) ? 2.0f : ((t == TT - 1) ? 0.0f : 1.0f);
        float m  = di - lo * cp_prev;
        float cp = up / m;
        alpha[t] = 1.0f / m;
        beta[t]  = -lo / m;
        gamma[t] = -cp;
        cp_prev  = cp;
    }
}

// ---------------------------------------------------------------------------
// K2: build per-chunk 16x16 scan matrices, pre-swizzled into the WMMA
// A-fragment layout for V_WMMA_F32_16X16X4_F32:
//   lane L holds M = L&15 ; for step q, VGPR v holds K = 4*q + v + 2*(L>>4)
// Stored so one lane's 8 floats per chunk are CONTIGUOUS (2x b128 loads):
//   index = (c*32 + L)*8 + q*2 + v
// Forward:  Lf[i][j] = alpha[t0+j] * prod_{k=j+1..i} beta[t0+k]   (j<=i)
//           gf[i]    = prod_{k=0..i} beta[t0+k]
// Backward: Lb[i][j] = prod_{k=i..j-1} gamma[t0+k]                (j>=i)
//           gb[i]    = prod_{k=i..15} gamma[t0+k]
// 64 blocks (one per chunk) x 256 threads: tid -> (q, L, v) exactly.
// ---------------------------------------------------------------------------
__global__ void tw_frag_kernel(float* ws) {
    const float* alpha = ws + WS_ALPHA;
    const float* beta  = ws + WS_BETA;
    const float* gamma = ws + WS_GAMMA;
    int c   = blockIdx.x;
    int tid = threadIdx.x;
    int q = tid >> 6;
    int L = (tid >> 1) & 31;
    int v = tid & 1;
    int t0 = c * 16;
    int i = L & 15;
    int j = 4 * q + v + 2 * (L >> 4);
    int slot = (c * 32 + L) * 8 + q * 2 + v;

    float fv;
    if (j > i) {
        fv = 0.0f;
    } else {
        fv = alpha[t0 + j];
        for (int k = j + 1; k <= i; ++k) fv *= beta[t0 + k];
    }
    ws[WS_AF + slot] = fv;

    float bv;
    if (j < i) {
        bv = 0.0f;
    } else {
        bv = 1.0f;
        for (int k = i; k < j; ++k) bv *= gamma[t0 + k];
    }
    ws[WS_AB + slot] = bv;

    if (tid < 16) {
        float g = 1.0f;
        for (int k = 0; k <= tid; ++k) g *= beta[t0 + k];
        ws[WS_GF + c * 16 + tid] = g;
        float g2 = 1.0f;
        for (int k = tid; k < 16; ++k) g2 *= gamma[t0 + k];
        ws[WS_GB + c * 16 + tid] = g2;
    }
}

// ---------------------------------------------------------------------------
// Spline-system RHS. Boundary version (only used in peeled chunks 0 / 63):
//   b[0]    = 0.5*(-5 y0 + 4 y1 + y2)
//   b[t]    = 3*(y[t+1] - y[t-1])              1 <= t <= T-2
//   b[T-1]  = 0.5*(-y[T-3] - 4 y[T-2] + 5 y[T-1])
// ---------------------------------------------------------------------------
__device__ __forceinline__ float tw_rhs_interior(const float* __restrict__ Y,
                                                 int t, int gcol) {
    float ym = Y[(long long)(t - 1) * NKL + gcol];
    float yp = Y[(long long)(t + 1) * NKL + gcol];
    return 3.0f * (yp - ym);
}

__device__ __forceinline__ float tw_rhs_boundary(const float* __restrict__ Y,
                                                 int t, int gcol) {
    if (t == 0) {
        float y0 = Y[gcol];
        float y1 = Y[NKL + gcol];
        float y2 = Y[2 * NKL + gcol];
        return 0.5f * (-5.0f * y0 + 4.0f * y1 + y2);
    }
    if (t == TT - 1) {
        float ya = Y[(long long)(TT - 3) * NKL + gcol];
        float yb = Y[(long long)(TT - 2) * NKL + gcol];
        float yc = Y[(long long)(TT - 1) * NKL + gcol];
        return 0.5f * (-ya - 4.0f * yb + 5.0f * yc);
    }
    return tw_rhs_interior(Y, t, gcol);
}

// One forward chunk: 4 WMMAs + carry add + LDS store. BOUNDARY selects the
// branchy RHS (chunks 0 and 63 only).
template <bool BOUNDARY>
__device__ __forceinline__ void tw_fwd_chunk(
    int c, int lane, int hi, int col16, int gcol,
    const float* __restrict__ Y, const float* __restrict__ Af,
    const float* __restrict__ GF, float* __restrict__ sstrip, float& dpin) {
    const int t0 = c * 16;
    const v2f* afrag = (const v2f*)(Af + (c * 32 + lane) * 8);
    v8f acc = {};
#pragma unroll
    for (int q = 0; q < 4; ++q) {
        v2f b;
        if (BOUNDARY) {
            b[0] = tw_rhs_boundary(Y, t0 + 4 * q + 0 + 2 * hi, gcol);
            b[1] = tw_rhs_boundary(Y, t0 + 4 * q + 1 + 2 * hi, gcol);
        } else {
            b[0] = tw_rhs_interior(Y, t0 + 4 * q + 0 + 2 * hi, gcol);
            b[1] = tw_rhs_interior(Y, t0 + 4 * q + 1 + 2 * hi, gcol);
        }
        acc = __builtin_amdgcn_wmma_f32_16x16x4_f32(
            false, afrag[q], false, b, (short)0, acc, false, false);
    }
    const float* gfc = GF + c * 16 + 8 * hi;
#pragma unroll
    for (int r = 0; r < 8; ++r) acc[r] += gfc[r] * dpin;
    // dp[t0+15] lives in VGPR7 of lanes 16..31 (M = 7+8, N = lane-16)
    float dplast = __shfl(acc[7], col16 + 16, 32);
#pragma unroll
    for (int r = 0; r < 8; ++r)
        sstrip[(t0 + r + 8 * hi) * 16 + col16] = acc[r];
    dpin = dplast;
}

// ---------------------------------------------------------------------------
// K3: fused scan + warp. One wave per 16-column tile of a single batch.
// ---------------------------------------------------------------------------
__global__ void __launch_bounds__(32)
tw_main_kernel(const float* __restrict__ Y, const float* __restrict__ WF,
               const float* __restrict__ ws, float* __restrict__ out) {
    __shared__ float sstrip[TT * 16];   // 64KB: dp, later s, for 16 columns

    const int lane  = threadIdx.x;      // 0..31 (wave32)
    const int hi    = lane >> 4;        // half-wave select
    const int col16 = lane & 15;        // local column / N index
    const int n     = blockIdx.x / 66;  // batch
    const int kt    = blockIdx.x % 66;  // 16-col tile within batch
    const int gcol  = n * KK + kt * 16 + col16;

    const float* Af = ws + WS_AF;
    const float* Ab = ws + WS_AB;
    const float* GF = ws + WS_GF;
    const float* GB = ws + WS_GB;

    // ---------------- Phase A: forward Thomas scan (chunked WMMA) ----------
    float dpin = 0.0f;
    tw_fwd_chunk<true>(0, lane, hi, col16, gcol, Y, Af, GF, sstrip, dpin);
    for (int c = 1; c < NCHUNK - 1; ++c) {
        // hint next chunk's rows toward the caches
        __builtin_prefetch(&Y[(long long)(c * 16 + 17) * NKL + gcol], 0, 1);
        tw_fwd_chunk<false>(c, lane, hi, col16, gcol, Y, Af, GF, sstrip, dpin);
    }
    tw_fwd_chunk<true>(NCHUNK - 1, lane, hi, col16, gcol, Y, Af, GF, sstrip, dpin);

    // ---------------- Phase B: backward substitution (reverse chunks) ------
    float xin = 0.0f;
    for (int c = NCHUNK - 1; c >= 0; --c) {
        const int t0 = c * 16;
        const v2f* afrag = (const v2f*)(Ab + (c * 32 + lane) * 8);
        v8f acc = {};
#pragma unroll
        for (int q = 0; q < 4; ++q) {
            v2f b;
            b[0] = sstrip[(t0 + 4 * q + 0 + 2 * hi) * 16 + col16];
            b[1] = sstrip[(t0 + 4 * q + 1 + 2 * hi) * 16 + col16];
            acc = __builtin_amdgcn_wmma_f32_16x16x4_f32(
                false, afrag[q], false, b, (short)0, acc, false, false);
        }
        const float* gbc = GB + c * 16 + 8 * hi;
#pragma unroll
        for (int r = 0; r < 8; ++r) acc[r] += gbc[r] * xin;
        // x[t0] lives in VGPR0 of lanes 0..15 (M = 0, N = lane)
        float xfirst = __shfl(acc[0], col16, 32);
#pragma unroll
        for (int r = 0; r < 8; ++r)
            sstrip[(t0 + r + 8 * hi) * 16 + col16] = acc[r];   // dp -> s in place
        xin = xfirst;
    }

    // ---------------- Phase C: Hermite warp for this batch -----------------
    const float factor = WF[n];
    const int   Lw     = (int)floorf((float)TT * factor);
    const float denom  = fmaxf((float)(Lw - 1), 1.0f);
    const float scale  = (float)(TT - 1) / denom;

    for (int it = 0; it < TT / 2; ++it) {
        const int j = 2 * it + hi;                 // output row (2 rows/iter)
        float tt = fminf((float)j * scale, (float)(TT - 1));
        int   i  = (int)tt;
        if (i > TT - 2) i = TT - 2;
        float u  = tt - (float)i;
        float y0 = Y[(long long)i * NKL + gcol];
        float y1 = Y[(long long)(i + 1) * NKL + gcol];
        float s0 = sstrip[i * 16 + col16];
        float s1 = sstrip[(i + 1) * 16 + col16];
        float u2 = u * u, u3 = u2 * u;
        float h00 =  2.0f * u3 - 3.0f * u2 + 1.0f;
        float h10 =         u3 - 2.0f * u2 + u;
        float h01 = -2.0f * u3 + 3.0f * u2;
        float h11 =         u3 -        u2;
        float val = h00 * y0 + h10 * s0 + h01 * y1 + h11 * s1;
        out[(long long)j * NKL + gcol] = (j < Lw) ? val : 0.0f;
    }
}

// ---------------------------------------------------------------------------
extern "C" void kernel_launch(void* const* d_in, const int* in_sizes, int n_in,
                              void* d_out, int out_size, void* d_ws, size_t ws_size,
                              hipStream_t stream) {
    (void)in_sizes; (void)n_in; (void)out_size; (void)ws_size;
    const float* Y  = (const float*)d_in[0];   // (T, N, BANDS, C, FREQ) f32
    const float* WF = (const float*)d_in[1];   // (N,) f32 warp factors
    float* out = (float*)d_out;
    float* ws  = (float*)d_ws;                 // needs ~152KB

    tw_coef_kernel<<<1, 1, 0, stream>>>(ws);
    tw_frag_kernel<<<NCHUNK, 256, 0, stream>>>(ws);
    tw_main_kernel<<<NN * 66, 32, 0, stream>>>(Y, WF, ws, out);
}